// RelationNetwork_78589311582322
// MI455X (gfx1250) — compile-verified
//
#include <hip/hip_runtime.h>
#include <math.h>

// Problem sizes (fixed by the reference)
#define BB  64      // batch
#define NN  2000    // nodes
#define RR  64      // rank (collapses analytically)
#define EMB 512
#define LAB 20

typedef __attribute__((ext_vector_type(2))) float v2f;
typedef __attribute__((ext_vector_type(4))) float v4f;
typedef __attribute__((ext_vector_type(8))) float v8f;

// Workspace layout (float offsets). Total = 288,960 floats (~1.13 MB).
static constexpr int WS_SCAL = 0;                 // [0]=a=dot(tw,pw), [1]=c2=dot(tb,pw)
static constexpr int WS_BMAX = 64;                // per-batch max(x)   (64)
static constexpr int WS_BMIN = 128;               // per-batch min(x)   (64)
static constexpr int WS_IGF  = 192;               // init_graph @ f^T, stored [n*64 + b]
static constexpr int WS_MIX  = WS_IGF + NN * BB;  // mixed [b*2000 + n] (even offset -> 8B aligned)
static constexpr int WS_HID  = WS_MIX + BB * NN;  // hidden [b*512 + e]

// ---------------------------------------------------------------------------
// Kernel 1: per-batch max/min of feature rows + the two collapsed scalars.
// ---------------------------------------------------------------------------
__global__ void prep_kernel(const float* __restrict__ f,
                            const float* __restrict__ tw,
                            const float* __restrict__ tb,
                            const float* __restrict__ pw,
                            float* __restrict__ W) {
  const int bid = blockIdx.x, tid = threadIdx.x;
  if (bid < BB) {
    __shared__ float smax[256], smin[256];
    float mx = -3.402823466e38f, mn = 3.402823466e38f;
    for (int i = tid; i < NN; i += 256) {
      float v = f[bid * NN + i];
      mx = fmaxf(mx, v);
      mn = fminf(mn, v);
    }
    smax[tid] = mx; smin[tid] = mn;
    __syncthreads();
    for (int s = 128; s > 0; s >>= 1) {
      if (tid < s) {
        smax[tid] = fmaxf(smax[tid], smax[tid + s]);
        smin[tid] = fminf(smin[tid], smin[tid + s]);
      }
      __syncthreads();
    }
    if (tid == 0) { W[WS_BMAX + bid] = smax[0]; W[WS_BMIN + bid] = smin[0]; }
  } else {
    // a = sum_r tw[r]*pw[r]; c2 = sum_r tb[r]*pw[r]  (R == 64)
    __shared__ float s1[64], s2[64];
    if (tid < RR) { float p = pw[tid]; s1[tid] = tw[tid] * p; s2[tid] = tb[tid] * p; }
    __syncthreads();
    for (int s = 32; s > 0; s >>= 1) {
      if (tid < s) { s1[tid] += s1[tid + s]; s2[tid] += s2[tid + s]; }
      __syncthreads();
    }
    if (tid == 0) { W[WS_SCAL + 0] = s1[0]; W[WS_SCAL + 1] = s2[0]; }
  }
}

// ---------------------------------------------------------------------------
// FP32 WMMA tile: C[16x16] = A[16xK] * Bt[16xK]^T using V_WMMA_F32_16X16X4_F32.
// "NT" layout: A element (m,k) at A[m*lda+k]; B element (k,n) at Bt[n*ldb+k].
// A-frag (ISA 7.12.2): lane l holds M=l&15, K = 2*(l>>4)+{vgpr0,vgpr1}.
// B-frag symmetric (N=l&15). C-frag: vgpr r -> M = r + 8*(l>>4), N = l&15.
// ---------------------------------------------------------------------------
__device__ __forceinline__ v8f wmma_tile_nt(const float* __restrict__ A, int lda,
                                            const float* __restrict__ Bt, int ldb,
                                            int m0, int n0, int K, int lane) {
  v8f acc = {};
  const int koff = (lane >> 4) * 2;                 // even -> float2 loads stay 8B aligned
  const float* ap = A  + (m0 + (lane & 15)) * lda + koff;
  const float* bp = Bt + (n0 + (lane & 15)) * ldb + koff;
  for (int k = 0; k < K; k += 4) {
    v2f a = *reinterpret_cast<const v2f*>(ap + k);
    v2f b = *reinterpret_cast<const v2f*>(bp + k);
    acc = __builtin_amdgcn_wmma_f32_16x16x4_f32(
        /*neg_a=*/false, a, /*neg_b=*/false, b,
        /*c_mod=*/(short)0, acc, /*reuse_a=*/false, /*reuse_b=*/false);
  }
  return acc;
}

// ---------------------------------------------------------------------------
// Kernel 2: igf[n,b] = sum_m init_graph[n,m] * f[b,m]   (2000x64, K=2000)
// 125 m-tiles x 4 b-tiles = 500 waves, one 16x16 C tile per wave.
// ---------------------------------------------------------------------------
__global__ void igf_gemm_kernel(const float* __restrict__ G,
                                const float* __restrict__ f,
                                float* __restrict__ W) {
  const int wave = (int)((blockIdx.x * blockDim.x + threadIdx.x) >> 5);
  if (wave >= 125 * 4) return;                      // wave-uniform exit; EXEC stays all-1s
  const int lane  = threadIdx.x & 31;
  const int m0 = (wave >> 2) * 16;                  // n-tile
  const int n0 = (wave & 3) * 16;                   // b-tile
  v8f acc = wmma_tile_nt(G, NN, f, NN, m0, n0, NN, lane);
  float* C = W + WS_IGF;
  const int row = m0 + (lane >> 4) * 8;
  const int col = n0 + (lane & 15);
#pragma unroll
  for (int r = 0; r < 8; ++r) C[(row + r) * BB + col] = acc[r];
}

// ---------------------------------------------------------------------------
// Kernel 3: collapsed softmax row + relu + residual (dominant kernel: ~256M exps).
// mixed[b,n] = relu( igf[n,b] + (Σ_m e(t x_m) x_m)/(Σ_m e(t x_m)) ) + x_n
//
// Hot-loop engineering:
//  * exp done in base-2 domain: t2 = t*log2(e) hoisted, so per element it is
//    exactly one v_fmac + one v_exp_f32 (no per-element ×log2e multiply).
//  * 2 output n's per thread -> each LDS broadcast feeds 2 exp chains.
//  * float4 LDS reads (ds_load_b128) -> 1 DS op per 8 exps.
// One block per (b, 512-chunk of n); x row staged in LDS.
// ---------------------------------------------------------------------------
__global__ void softmax_kernel(const float* __restrict__ f,
                               float* __restrict__ W) {
  __shared__ float sx[NN];                          // 8000 B of the 320 KB LDS
  const int b    = blockIdx.x >> 2;
  const int tile = blockIdx.x & 3;
  const int tid  = threadIdx.x;
  const float* xrow = f + b * NN;
  for (int i = tid; i < NN; i += 256) sx[i] = xrow[i];
  __syncthreads();

  const float L2E = 1.4426950408889634f;
  const float a   = W[WS_SCAL + 0];
  const float c2  = W[WS_SCAL + 1];
  const float bmx = W[WS_BMAX + b];
  const float bmn = W[WS_BMIN + b];

  const int n0 = tile * 512 + tid;                  // always < 2000
  const int n1 = n0 + 256;                          // may exceed 1999 on last tile
  const float xn0 = sx[n0];
  const float xn1 = sx[(n1 < NN) ? n1 : (NN - 1)];

  // base-2 domain slopes and (negated) row maxima
  const float t20  = (a * xn0 + c2) * L2E;
  const float t21  = (a * xn1 + c2) * L2E;
  const float nmx0 = -((t20 >= 0.f) ? t20 * bmx : t20 * bmn);
  const float nmx1 = -((t21 >= 0.f) ? t21 * bmx : t21 * bmn);

  float den0 = 0.f, num0 = 0.f, den1 = 0.f, num1 = 0.f;
  for (int m = 0; m < NN; m += 4) {
    const v4f xv = *reinterpret_cast<const v4f*>(&sx[m]);   // ds_load_b128 broadcast
#pragma unroll
    for (int j = 0; j < 4; ++j) {
      const float xm = xv[j];
      const float e0 = __builtin_amdgcn_exp2f(fmaf(t20, xm, nmx0));  // v_exp_f32
      const float e1 = __builtin_amdgcn_exp2f(fmaf(t21, xm, nmx1));
      den0 += e0; num0 = fmaf(e0, xm, num0);
      den1 += e1; num1 = fmaf(e1, xm, num1);
    }
  }

  {
    const float rel = num0 / den0 + W[WS_IGF + n0 * BB + b];
    W[WS_MIX + b * NN + n0] = fmaxf(rel, 0.f) + xn0;
  }
  if (n1 < NN) {
    const float rel = num1 / den1 + W[WS_IGF + n1 * BB + b];
    W[WS_MIX + b * NN + n1] = fmaxf(rel, 0.f) + xn1;
  }
}

// ---------------------------------------------------------------------------
// Kernel 4: hidden = tanh(mixed @ fc1_w^T + fc1_b)  (64x512, K=2000), WMMA.
// ---------------------------------------------------------------------------
__global__ void fc1_gemm_kernel(const float* __restrict__ fc1w,
                                const float* __restrict__ fc1b,
                                float* __restrict__ W) {
  const int wave = (int)((blockIdx.x * blockDim.x + threadIdx.x) >> 5);
  if (wave >= 4 * 32) return;                       // 4 b-tiles x 32 e-tiles
  const int lane = threadIdx.x & 31;
  const int m0 = (wave >> 5) * 16;                  // batch tile
  const int n0 = (wave & 31) * 16;                  // embed tile
  v8f acc = wmma_tile_nt(W + WS_MIX, NN, fc1w, NN, m0, n0, NN, lane);
  const float bias = fc1b[n0 + (lane & 15)];
  float* C = W + WS_HID;
  const int row = m0 + (lane >> 4) * 8;
  const int col = n0 + (lane & 15);
#pragma unroll
  for (int r = 0; r < 8; ++r) C[(row + r) * EMB + col] = tanhf(acc[r] + bias);
}

// ---------------------------------------------------------------------------
// Kernel 5: logits = hidden @ fc2_w^T + fc2_b  (64x20, K=512) — tiny, plain VALU.
// ---------------------------------------------------------------------------
__global__ void fc2_kernel(const float* __restrict__ fc2w,
                           const float* __restrict__ fc2b,
                           const float* __restrict__ W,
                           float* __restrict__ out) {
  const int idx = blockIdx.x * blockDim.x + threadIdx.x;
  if (idx >= BB * LAB) return;
  const int b = idx / LAB, l = idx % LAB;
  const float* h  = W + WS_HID + b * EMB;
  const float* wv = fc2w + l * EMB;
  float s = fc2b[l];
#pragma unroll 4
  for (int e = 0; e < EMB; ++e) s = fmaf(h[e], wv[e], s);
  out[idx] = s;
}

// ---------------------------------------------------------------------------
extern "C" void kernel_launch(void* const* d_in, const int* in_sizes, int n_in,
                              void* d_out, int out_size, void* d_ws, size_t ws_size,
                              hipStream_t stream) {
  (void)in_sizes; (void)n_in; (void)out_size; (void)ws_size;
  const float* f    = (const float*)d_in[0];  // feature   [64,2000]
  const float* G    = (const float*)d_in[1];  // init_graph[2000,2000]
  const float* tw   = (const float*)d_in[2];  // theta_w   [64,1]
  const float* tb   = (const float*)d_in[3];  // theta_b   [64]
  const float* pw   = (const float*)d_in[4];  // phi_w     [64,1]
  /* d_in[5] = phi_b cancels inside the softmax (m-independent term) */
  const float* fc1w = (const float*)d_in[6];  // [512,2000]
  const float* fc1b = (const float*)d_in[7];  // [512]
  const float* fc2w = (const float*)d_in[8];  // [20,512]
  const float* fc2b = (const float*)d_in[9];  // [20]
  float* out = (float*)d_out;                 // logits [64,20]
  float* W   = (float*)d_ws;

  prep_kernel<<<BB + 1, 256, 0, stream>>>(f, tw, tb, pw, W);
  igf_gemm_kernel<<<(500 + 7) / 8, 256, 0, stream>>>(G, f, W);     // 500 waves, 8/block
  softmax_kernel<<<BB * 4, 256, 0, stream>>>(f, W);                // 2 n per thread
  fc1_gemm_kernel<<<128 / 8, 256, 0, stream>>>(fc1w, fc1b, W);     // 128 waves
  fc2_kernel<<<(BB * LAB + 255) / 256, 256, 0, stream>>>(fc2w, fc2b, W, out);
}